// ExpertChoiceRouter_29016799052558
// MI455X (gfx1250) — compile-verified
//
#include <hip/hip_runtime.h>

typedef __attribute__((ext_vector_type(2))) float v2f;
typedef __attribute__((ext_vector_type(8))) float v8f;

#define DMODEL 1024
#define SEQ    8192
#define NBATCH 8

// ---------------------------------------------------------------------------
// Kernel 1: scores[b,s] = dot(H[b,s,:], w) + bias, via V_WMMA_F32_16X16X4_F32.
// Each wave computes 16 token scores. A-matrix (16x4 f32, 2 VGPRs):
//   lanes 0-15:  M=lane,    K=0 (vgpr0), K=1 (vgpr1)
//   lanes 16-31: M=lane-16, K=2 (vgpr0), K=3 (vgpr1)
// -> per-lane aligned float2 load at H[tok]*D + d0 + 2*(lane>>4).
// B-matrix (4x16) holds w[d0+k] replicated across all 16 columns, so the
// value depends only on K and any row-striping order is equivalent; we mirror
// the A-side K layout. C[m][n] is identical for all n; C vgpr r holds token
// r (lanes 0-15) and token r+8 (lanes 16-31).
// ---------------------------------------------------------------------------
__global__ __launch_bounds__(256) void gemv_wmma_f32(
    const float* __restrict__ H, const float* __restrict__ w,
    const float* __restrict__ bias, float* __restrict__ scores) {
  const int lane    = threadIdx.x & 31;
  const int waveId  = (int)((blockIdx.x * blockDim.x + threadIdx.x) >> 5);
  const int tokBase = waveId * 16;
  const int m       = lane & 15;
  const int g       = lane >> 4;                       // 0: K=0,1  1: K=2,3
  const float* rowp = H + (size_t)(tokBase + m) * DMODEL + 2 * g;
  const float* wp   = w + 2 * g;

  v8f c = {};
#pragma unroll 4
  for (int d0 = 0; d0 < DMODEL; d0 += 4) {
    v2f a  = *(const v2f*)(rowp + d0);                 // 8B aligned
    v2f bb = *(const v2f*)(wp + d0);
    // (neg_a, A, neg_b, B, c_mod, C, reuse_a, reuse_b)
    c = __builtin_amdgcn_wmma_f32_16x16x4_f32(false, a, false, bb,
                                              (short)0, c, false, false);
  }
  const float bv = bias[0];
  if ((lane & 15) == 0) {                              // lanes 0 and 16
    float* outp = scores + tokBase + 8 * g;
#pragma unroll
    for (int r = 0; r < 8; ++r) outp[r] = c[r] + bv;
  }
}

// ---------------------------------------------------------------------------
// Kernel 2: per-row variable-k top-k with stable (earliest-index) tie break.
// One block (1024 threads) per batch row; keys live in LDS.
// ---------------------------------------------------------------------------
__global__ __launch_bounds__(1024) void topk_select(
    const float* __restrict__ scores, const int* __restrict__ mask,
    float* __restrict__ out) {
  __shared__ unsigned keys[SEQ];        // 32 KB
  __shared__ unsigned scanb[1024];
  __shared__ unsigned cnt[40];          // pre-zeroed reduction slots

  const int b   = blockIdx.x;
  const int tid = threadIdx.x;
  if (tid < 40) cnt[tid] = 0u;
  __syncthreads();

  // Load row: monotone key (larger score -> larger key); inactive -> 0.
  unsigned nloc = 0;
  for (int s = tid; s < SEQ; s += 1024) {
    float    v    = scores[b * SEQ + s];
    unsigned bits = __float_as_uint(v);
    unsigned key  = bits ^ (unsigned)(((int)bits >> 31) | 0x80000000);
    bool     a    = mask[b * SEQ + s] != 0;
    keys[s] = a ? key : 0u;
    nloc   += a ? 1u : 0u;
  }
  for (int off = 16; off > 0; off >>= 1) nloc += __shfl_down(nloc, off, 32);
  if ((tid & 31) == 0) atomicAdd(&cnt[0], nloc);
  __syncthreads();
  const unsigned n_active = cnt[0];

  int kc = (int)ceilf((float)n_active * 0.5f);
  if (kc < 1) kc = 1;
  const unsigned k = (unsigned)kc;

  // MSB radix-select of the k-th largest key.
  unsigned prefix = 0, kk = k;
  for (int bit = 31; bit >= 0; --bit) {
    const unsigned candHigh = (prefix >> bit) | 1u;
    unsigned c = 0;
    for (int s = tid; s < SEQ; s += 1024)
      c += ((keys[s] >> bit) == candHigh) ? 1u : 0u;
    for (int off = 16; off > 0; off >>= 1) c += __shfl_down(c, off, 32);
    const int slot = 1 + (31 - bit);
    if ((tid & 31) == 0) atomicAdd(&cnt[slot], c);
    __syncthreads();
    const unsigned total = cnt[slot];
    if (total >= kk) prefix |= (1u << bit);
    else             kk -= total;
  }
  const unsigned T = prefix;   // k-th largest key
  const unsigned E = kk;       // #ties (key==T) to keep, earliest-index first

  // Stable tie ranking: exclusive scan of tie counts over contiguous chunks.
  const int base = tid * 8;
  unsigned tc = 0;
#pragma unroll
  for (int j = 0; j < 8; ++j) tc += (keys[base + j] == T) ? 1u : 0u;
  scanb[tid] = tc;
  __syncthreads();
  for (int off = 1; off < 1024; off <<= 1) {
    unsigned v = (tid >= off) ? scanb[tid - off] : 0u;
    __syncthreads();
    scanb[tid] += v;
    __syncthreads();
  }
  unsigned run = scanb[tid] - tc;  // exclusive prefix of ties before my chunk

#pragma unroll
  for (int j = 0; j < 8; ++j) {
    const unsigned key = keys[base + j];
    bool keep;
    if (key > T)        keep = true;
    else if (key == T) { keep = (run < E); run++; }
    else                keep = false;
    keep = keep && (key != 0u);              // AND active_mask
    out[b * SEQ + base + j] = keep ? 1.0f : 0.0f;
  }
}

extern "C" void kernel_launch(void* const* d_in, const int* in_sizes, int n_in,
                              void* d_out, int out_size, void* d_ws, size_t ws_size,
                              hipStream_t stream) {
  const float* H    = (const float*)d_in[0];
  const int*   mask = (const int*)d_in[1];
  const float* w    = (const float*)d_in[2];
  const float* bias = (const float*)d_in[3];
  float*       out  = (float*)d_out;

  const size_t scoreBytes = (size_t)NBATCH * SEQ * sizeof(float);
  float* scores = (ws_size >= scoreBytes) ? (float*)d_ws : out;

  // 65536 tokens / (8 waves * 16 tokens) = 512 blocks of 256 threads
  gemv_wmma_f32<<<(NBATCH * SEQ) / 128, 256, 0, stream>>>(H, w, bias, scores);
  topk_select<<<NBATCH, 1024, 0, stream>>>(scores, mask, out);
}